// GNSMsg_EdgeSelfAttn_33260226740748
// MI455X (gfx1250) — compile-verified
//
#include <hip/hip_runtime.h>
#include <hip/hip_bf16.h>
#include <math.h>

typedef float v2f __attribute__((ext_vector_type(2)));
typedef float v8f __attribute__((ext_vector_type(8)));

#define LN_EPS 1e-5f

// D = A(16x4,f32) * B(4x16,f32) + C(16x16,f32) -- exact fp32 matrix op on CDNA5.
__device__ __forceinline__ v8f wmma4(v2f a, v2f b, v8f c) {
    return __builtin_amdgcn_wmma_f32_16x16x4_f32(
        /*neg_a=*/false, a, /*neg_b=*/false, b,
        /*c_mod=*/(short)0, c, /*reuse_a=*/false, /*reuse_b=*/false);
}

// Monotone float->uint encoding so unsigned atomicMax == float max (exact, deterministic).
__device__ __forceinline__ unsigned encf(float f) {
    unsigned b = __float_as_uint(f);
    return (b & 0x80000000u) ? ~b : (b | 0x80000000u);
}
__device__ __forceinline__ float decf(unsigned u) {
    unsigned b = (u & 0x80000000u) ? (u & 0x7FFFFFFFu) : ~u;
    return __uint_as_float(b);
}

// Cooperative 16x64 f32 tile load: global (row-major, stride 64) -> LDS, b128 per lane.
__device__ __forceinline__ void load_tile16x64(float* lds, const float* __restrict__ src,
                                               int node0, int N, int lane, bool full) {
#pragma unroll
    for (int it = 0; it < 8; ++it) {
        int c = lane + 32 * it;          // float4-chunk id, 16 chunks per row
        int row = c >> 4;
        int gm = node0 + row;
        float4 v;
        if (full || gm < N) {
            v = *(const float4*)&src[(size_t)gm * 64 + (c & 15) * 4];
        } else {
            v = make_float4(0.f, 0.f, 0.f, 0.f);
        }
        *(float4*)&lds[c * 4] = v;
    }
}

// ---------------------------------------------------------------------------
// Kernel 1: LN1 + Q/K/V projections. One wave (32 threads) per 16-node tile.
// ---------------------------------------------------------------------------
__global__ void k_ln_qkv(const float* __restrict__ x,
                         const float* __restrict__ g, const float* __restrict__ bb_,
                         const float* __restrict__ Wq, const float* __restrict__ Wk,
                         const float* __restrict__ Wv,
                         float* __restrict__ Q, float* __restrict__ K, float* __restrict__ V,
                         int N) {
    __shared__ float y[16 * 64];
    const int lane = threadIdx.x;
    const int half = lane >> 4;
    const int l    = lane & 15;
    const int node0 = blockIdx.x * 16;
    const bool full = (node0 + 16 <= N);

    load_tile16x64(y, x, node0, N, lane, full);
    __syncthreads();

    // LayerNorm in place (each of 16 lanes owns one row, reads/writes LDS).
    if (lane < 16) {
        float mu = 0.f;
        for (int d = 0; d < 64; ++d) mu += y[lane * 64 + d];
        mu *= (1.f / 64.f);
        float var = 0.f;
        for (int d = 0; d < 64; ++d) { float t = y[lane * 64 + d] - mu; var += t * t; }
        var *= (1.f / 64.f);
        float r = rsqrtf(var + LN_EPS);
        for (int d = 0; d < 64; ++d)
            y[lane * 64 + d] = (y[lane * 64 + d] - mu) * r * g[d] + bb_[d];
    }
    __syncthreads();

    // A fragments: lane l holds row M=l; half selects K offset {0,1} vs {2,3}.
    v2f afrag[16];
#pragma unroll
    for (int kk = 0; kk < 16; ++kk)
        afrag[kk] = *(const v2f*)&y[l * 64 + kk * 4 + half * 2];

    const float* Ws[3] = {Wq, Wk, Wv};
    float* Os[3] = {Q, K, V};
#pragma unroll
    for (int w = 0; w < 3; ++w) {
        const float* W = Ws[w];
        float* O = Os[w];
#pragma unroll
        for (int nt = 0; nt < 4; ++nt) {
            v8f acc = {};
            const int n = nt * 16 + l;
#pragma unroll
            for (int kk = 0; kk < 16; ++kk) {
                const int kbase = kk * 4 + half * 2;
                v2f bv;
                bv.x = W[(size_t)kbase * 64 + n];
                bv.y = W[(size_t)(kbase + 1) * 64 + n];
                acc = wmma4(afrag[kk], bv, acc);
            }
            if (full) {
#pragma unroll
                for (int i = 0; i < 8; ++i)
                    O[(size_t)(node0 + i + half * 8) * 64 + n] = acc[i];
            } else {
#pragma unroll
                for (int i = 0; i < 8; ++i) {
                    int gm = node0 + i + half * 8;
                    if (gm < N) O[(size_t)gm * 64 + n] = acc[i];
                }
            }
        }
    }
}

// ---------------------------------------------------------------------------
// Kernel 2: edge bias MLP + per-head logits + segment max (encoded atomicMax).
// ---------------------------------------------------------------------------
__global__ void k_edge_logits(const int* __restrict__ ei, const float* __restrict__ ef,
                              const float* __restrict__ Q, const float* __restrict__ K,
                              const float* __restrict__ w1, const float* __restrict__ b1,
                              const float* __restrict__ w2, const float* __restrict__ b2,
                              float* __restrict__ logits, unsigned* __restrict__ mxu,
                              int E) {
    int e = blockIdx.x * blockDim.x + threadIdx.x;
    if (e >= E) return;
    const int src = ei[2 * e + 0];
    const int dst = ei[2 * e + 1];

    float f[9];
#pragma unroll
    for (int i = 0; i < 9; ++i) f[i] = ef[(size_t)e * 9 + i];

    float h1[18];
#pragma unroll
    for (int j = 0; j < 18; ++j) {
        float s = b1[j];
#pragma unroll
        for (int i = 0; i < 9; ++i) s += f[i] * w1[i * 18 + j];
        h1[j] = (s > 0.f) ? s : 0.1f * s;  // leaky_relu 0.1
    }
    float bias[4];
#pragma unroll
    for (int h = 0; h < 4; ++h) {
        float s = b2[h];
#pragma unroll
        for (int j = 0; j < 18; ++j) s += h1[j] * w2[j * 4 + h];
        bias[h] = s;
    }

    const float* q = Q + (size_t)dst * 64;
    const float* k = K + (size_t)src * 64;
#pragma unroll
    for (int h = 0; h < 4; ++h) {
        float s = 0.f;
#pragma unroll
        for (int i = 0; i < 16; ++i) s += q[h * 16 + i] * k[h * 16 + i];
        float lg = s * 0.25f + bias[h];  // / sqrt(16)
        logits[(size_t)e * 4 + h] = lg;
        atomicMax(&mxu[(size_t)dst * 4 + h], encf(lg));
    }
}

// ---------------------------------------------------------------------------
// Kernel 3: ex = exp(logit - max); atomic fadd into denominator.
// ---------------------------------------------------------------------------
__global__ void k_edge_exp(const int* __restrict__ ei, float* __restrict__ logits,
                           const unsigned* __restrict__ mxu, float* __restrict__ den,
                           int E4) {
    int t = blockIdx.x * blockDim.x + threadIdx.x;
    if (t >= E4) return;
    int e = t >> 2, h = t & 3;
    int dst = ei[2 * e + 1];
    float mx = decf(mxu[(size_t)dst * 4 + h]);
    float ex = expf(logits[t] - mx);
    logits[t] = ex;
    __hip_atomic_fetch_add(&den[(size_t)dst * 4 + h], ex,
                           __ATOMIC_RELAXED, __HIP_MEMORY_SCOPE_AGENT);
}

// ---------------------------------------------------------------------------
// Kernel 4: msg = alpha * Vj scatter-add into agg. Thread per (edge, float4).
// ---------------------------------------------------------------------------
__global__ void k_edge_msg(const int* __restrict__ ei, const float* __restrict__ ex_,
                           const float* __restrict__ den, const float* __restrict__ V,
                           float* __restrict__ agg, int total) {
    int t = blockIdx.x * blockDim.x + threadIdx.x;
    if (t >= total) return;
    int e  = t >> 4;
    int qi = t & 15;          // which float4 of the 64-dim row
    int i0 = qi * 4;
    int h  = qi >> 2;         // head = i0/16
    int src = ei[2 * e + 0];
    int dst = ei[2 * e + 1];
    float alpha = ex_[(size_t)e * 4 + h] / (den[(size_t)dst * 4 + h] + 1e-12f);
    float4 v4 = *(const float4*)&V[(size_t)src * 64 + i0];
    float* ap = &agg[(size_t)dst * 64 + i0];
    __hip_atomic_fetch_add(ap + 0, alpha * v4.x, __ATOMIC_RELAXED, __HIP_MEMORY_SCOPE_AGENT);
    __hip_atomic_fetch_add(ap + 1, alpha * v4.y, __ATOMIC_RELAXED, __HIP_MEMORY_SCOPE_AGENT);
    __hip_atomic_fetch_add(ap + 2, alpha * v4.z, __ATOMIC_RELAXED, __HIP_MEMORY_SCOPE_AGENT);
    __hip_atomic_fetch_add(ap + 3, alpha * v4.w, __ATOMIC_RELAXED, __HIP_MEMORY_SCOPE_AGENT);
}

// ---------------------------------------------------------------------------
// Kernel 5: out-projection + residual + LN2 + FFN + residual. One wave / tile.
// ---------------------------------------------------------------------------
__global__ void k_out_ffn(const float* __restrict__ x, const float* __restrict__ agg,
                          const float* __restrict__ Wo,
                          const float* __restrict__ ln2g, const float* __restrict__ ln2b,
                          const float* __restrict__ fw1, const float* __restrict__ fb1,
                          const float* __restrict__ fw2, const float* __restrict__ fb2,
                          float* __restrict__ out, int N) {
    __shared__ float x2[16 * 64];
    __shared__ float z[16 * 64];
    __shared__ float hbuf[16 * 256];
    const int lane = threadIdx.x;
    const int half = lane >> 4;
    const int l    = lane & 15;
    const int node0 = blockIdx.x * 16;
    const bool full = (node0 + 16 <= N);

    // Stage the x residual tile in LDS (coalesced b128 loads).
    load_tile16x64(x2, x, node0, N, lane, full);

    // A fragments from agg tile (b64 vector loads).
    v2f afrag[16];
    {
        int gm = node0 + l;
        if (full || gm < N) {
#pragma unroll
            for (int kk = 0; kk < 16; ++kk)
                afrag[kk] = *(const v2f*)&agg[(size_t)gm * 64 + kk * 4 + half * 2];
        } else {
#pragma unroll
            for (int kk = 0; kk < 16; ++kk) { afrag[kk].x = 0.f; afrag[kk].y = 0.f; }
        }
    }
    __syncthreads();

    // agg @ Wo + x -> x2 (LDS, in place)
#pragma unroll
    for (int nt = 0; nt < 4; ++nt) {
        v8f acc = {};
        const int n = nt * 16 + l;
#pragma unroll
        for (int kk = 0; kk < 16; ++kk) {
            const int kbase = kk * 4 + half * 2;
            v2f bv;
            bv.x = Wo[(size_t)kbase * 64 + n];
            bv.y = Wo[(size_t)(kbase + 1) * 64 + n];
            acc = wmma4(afrag[kk], bv, acc);
        }
#pragma unroll
        for (int i = 0; i < 8; ++i) {
            int m = i + half * 8;
            x2[m * 64 + n] += acc[i];
        }
    }
    __syncthreads();

    // LN2 per row
    if (lane < 16) {
        float mu = 0.f;
        for (int d = 0; d < 64; ++d) mu += x2[lane * 64 + d];
        mu *= (1.f / 64.f);
        float var = 0.f;
        for (int d = 0; d < 64; ++d) { float t = x2[lane * 64 + d] - mu; var += t * t; }
        var *= (1.f / 64.f);
        float r = rsqrtf(var + LN_EPS);
        for (int d = 0; d < 64; ++d)
            z[lane * 64 + d] = (x2[lane * 64 + d] - mu) * r * ln2g[d] + ln2b[d];
    }
    __syncthreads();

    // z(16x64) @ fw1(64x256) + b1, exact gelu -> hbuf
#pragma unroll
    for (int nt = 0; nt < 16; ++nt) {
        v8f acc = {};
        const int n = nt * 16 + l;
#pragma unroll
        for (int kk = 0; kk < 16; ++kk) {
            v2f a = *(const v2f*)&z[l * 64 + kk * 4 + half * 2];
            const int kbase = kk * 4 + half * 2;
            v2f bv;
            bv.x = fw1[(size_t)kbase * 256 + n];
            bv.y = fw1[(size_t)(kbase + 1) * 256 + n];
            acc = wmma4(a, bv, acc);
        }
#pragma unroll
        for (int i = 0; i < 8; ++i) {
            int m = i + half * 8;
            float v = acc[i] + fb1[n];
            hbuf[m * 256 + n] = 0.5f * v * (1.0f + erff(v * 0.70710678118654752f));
        }
    }
    __syncthreads();

    // hbuf(16x256) @ fw2(256x64) + b2 + x2 -> out
#pragma unroll
    for (int nt = 0; nt < 4; ++nt) {
        v8f acc = {};
        const int n = nt * 16 + l;
        for (int kk = 0; kk < 64; ++kk) {
            v2f a = *(const v2f*)&hbuf[l * 256 + kk * 4 + half * 2];
            const int kbase = kk * 4 + half * 2;
            v2f bv;
            bv.x = fw2[(size_t)kbase * 64 + n];
            bv.y = fw2[(size_t)(kbase + 1) * 64 + n];
            acc = wmma4(a, bv, acc);
        }
        if (full) {
#pragma unroll
            for (int i = 0; i < 8; ++i) {
                int m = i + half * 8;
                out[(size_t)(node0 + m) * 64 + n] = x2[m * 64 + n] + acc[i] + fb2[n];
            }
        } else {
#pragma unroll
            for (int i = 0; i < 8; ++i) {
                int m = i + half * 8;
                int gm = node0 + m;
                if (gm < N) out[(size_t)gm * 64 + n] = x2[m * 64 + n] + acc[i] + fb2[n];
            }
        }
    }
}

// ---------------------------------------------------------------------------
extern "C" void kernel_launch(void* const* d_in, const int* in_sizes, int n_in,
                              void* d_out, int out_size, void* d_ws, size_t ws_size,
                              hipStream_t stream) {
    const float* x    = (const float*)d_in[0];
    const int*   ei   = (const int*)d_in[1];
    const float* ef   = (const float*)d_in[2];
    const float* Wq   = (const float*)d_in[3];
    const float* Wk   = (const float*)d_in[4];
    const float* Wv   = (const float*)d_in[5];
    const float* Wo   = (const float*)d_in[6];
    const float* ebw1 = (const float*)d_in[7];
    const float* ebb1 = (const float*)d_in[8];
    const float* ebw2 = (const float*)d_in[9];
    const float* ebb2 = (const float*)d_in[10];
    const float* ln1g = (const float*)d_in[11];
    const float* ln1b = (const float*)d_in[12];
    const float* ln2g = (const float*)d_in[13];
    const float* ln2b = (const float*)d_in[14];
    const float* fw1  = (const float*)d_in[15];
    const float* fb1  = (const float*)d_in[16];
    const float* fw2  = (const float*)d_in[17];
    const float* fb2  = (const float*)d_in[18];

    const int N = in_sizes[0] / 64;
    const int E = in_sizes[1] / 2;

    float* ws = (float*)d_ws;
    float* Q      = ws;
    float* K      = Q + (size_t)N * 64;
    float* V      = K + (size_t)N * 64;
    float* logits = V + (size_t)N * 64;                 // E*4, becomes ex after k_edge_exp
    unsigned* mxu = (unsigned*)(logits + (size_t)E * 4);
    float* den    = (float*)(mxu + (size_t)N * 4);
    float* agg    = den + (size_t)N * 4;

    // mx (encoded: 0 == "-inf"), den, agg are contiguous -> one async memset.
    size_t clear_bytes = ((size_t)N * 4 + (size_t)N * 4 + (size_t)N * 64) * sizeof(float);
    hipMemsetAsync(mxu, 0, clear_bytes, stream);

    const int ntile = (N + 15) / 16;

    k_ln_qkv<<<ntile, 32, 0, stream>>>(x, ln1g, ln1b, Wq, Wk, Wv, Q, K, V, N);

    k_edge_logits<<<(E + 255) / 256, 256, 0, stream>>>(ei, ef, Q, K, ebw1, ebb1, ebw2, ebb2,
                                                       logits, mxu, E);

    k_edge_exp<<<(E * 4 + 255) / 256, 256, 0, stream>>>(ei, logits, mxu, den, E * 4);

    k_edge_msg<<<(E * 16 + 255) / 256, 256, 0, stream>>>(ei, logits, den, V, agg, E * 16);

    k_out_ffn<<<ntile, 32, 0, stream>>>(x, agg, Wo, ln2g, ln2b, fw1, fb1, fw2, fb2,
                                        (float*)d_out, N);
}